// WaveRNN_77927886618842
// MI455X (gfx1250) — compile-verified
//
#include <hip/hip_runtime.h>

// Persistent-kernel 2D acoustic FDTD for MI455X (gfx1250, wave32).
// 8 shots x (256x256) x 500 steps. Each shot split across 2 workgroups
// (128 rows each); p_cur lives in LDS, p_prev/c2/p_next in registers.
// Inter-WG halo rows exchanged via L2 with double-buffered rows +
// release/acquire flag handshake; halo receive uses the CDNA5 async
// global->LDS path (ASYNCcnt) when the builtin is available.

#define NT    500
#define NS    8
#define NZ    256
#define NXX   256
#define NREC  128
#define HROWS 128           // rows owned per workgroup
#define LROWS 130           // + top/bottom halo rows
#define STR   264           // padded LDS row stride (floats)
#define COFF  4             // global col c -> LDS col index c+4 (16B aligned)

typedef int v4i __attribute__((ext_vector_type(4)));
typedef __attribute__((address_space(1))) v4i* gv4i_ptr;   // global (AS1)
typedef __attribute__((address_space(3))) v4i* lv4i_ptr;   // LDS (AS3)

#if defined(__has_builtin)
#  if __has_builtin(__builtin_amdgcn_global_load_async_to_lds_b128)
#    define HAVE_ASYNC_LDS 1
#  endif
#endif
#ifndef HAVE_ASYNC_LDS
#  define HAVE_ASYNC_LDS 0
#endif

__global__ void wave_init_flags(int* flags) {
    if (threadIdx.x < NS * 2) flags[threadIdx.x] = 0;
}

__global__ __launch_bounds__(1024, 1)
void wave_persistent(const float* __restrict__ x,      // (NT, NS)
                     const float* __restrict__ vp,     // (NZ, NX)
                     const int*   __restrict__ src,    // (NS, 2)
                     const int*   __restrict__ rec,    // (NS, NREC, 2)
                     float*       __restrict__ out,    // (NT, NS, NREC)
                     float*       __restrict__ halo,   // (NS, 2 sender, 2 parity, NX)
                     int*         __restrict__ flags)  // (NS, 2)
{
    extern __shared__ float P[];                 // LROWS x STR floats (~134 KB)
    const int tid  = threadIdx.x;
    const int shot = blockIdx.x >> 1;
    const int me   = blockIdx.x & 1;             // 0 = rows 0..127, 1 = rows 128..255
    const int row0 = me * HROWS;
    const int ty   = tid >> 5;                   // 0..31 (one wave per ty)
    const int tx   = tid & 31;                   // 0..31
    const int lr0  = ty * 4 + 1;                 // first owned local row (1..128)
    const int gc0  = tx * 8;                     // first owned global col
    const int lc0  = gc0 + COFF;                 // LDS col index (16B aligned)

    // Zero the whole LDS tile (halo rows/cols stay zero forever -> zero padding).
    for (int i = tid; i < LROWS * STR; i += 1024) P[i] = 0.0f;

    // c2 = (vp * DT/DX)^2 into registers; p_prev starts at zero.
    float c2r[4][8], pp[4][8];
    #pragma unroll
    for (int i = 0; i < 4; ++i) {
        const float* vrow = vp + (row0 + ty * 4 + i) * NXX + gc0;
        #pragma unroll
        for (int j = 0; j < 8; ++j) {
            float s = vrow[j] * 1.0e-4f;         // DT/DX = 0.001/10
            c2r[i][j] = s * s;
            pp[i][j]  = 0.0f;
        }
    }

    // Receiver coords (first 128 threads own one receiver each).
    int rz = -1, rx = 0;
    if (tid < NREC) {
        rz = rec[(shot * NREC + tid) * 2 + 0];
        rx = rec[(shot * NREC + tid) * 2 + 1];
    }
    const int sz = src[shot * 2 + 0];
    const int sx = src[shot * 2 + 1];
    const bool own_src = (sz >= row0) && (sz < row0 + HROWS);

    const int sendlr = (me == 0) ? HROWS     : 1;    // boundary row we export
    const int recvlr = (me == 0) ? HROWS + 1 : 0;    // halo row we import
    int* const myflag = &flags[shot * 2 + me];
    int* const pflag  = &flags[shot * 2 + (1 - me)];

    __syncthreads();

    for (int t = 0; t < NT; ++t) {
        // ---- compute p_next from LDS p_cur + register p_prev ----
        float pcv[6][8], pn[4][8];
        #pragma unroll
        for (int r = 0; r < 6; ++r) {
            const float* prow = &P[(lr0 - 1 + r) * STR + lc0];
            float4 a = *(const float4*)(prow);
            float4 b = *(const float4*)(prow + 4);
            pcv[r][0] = a.x; pcv[r][1] = a.y; pcv[r][2] = a.z; pcv[r][3] = a.w;
            pcv[r][4] = b.x; pcv[r][5] = b.y; pcv[r][6] = b.z; pcv[r][7] = b.w;
        }
        #pragma unroll
        for (int i = 0; i < 4; ++i) {
            const float pl = P[(lr0 + i) * STR + lc0 - 1];
            const float pr = P[(lr0 + i) * STR + lc0 + 8];
            #pragma unroll
            for (int j = 0; j < 8; ++j) {
                const float c   = pcv[i + 1][j];
                const float lft = (j == 0) ? pl : pcv[i + 1][j - 1];
                const float rgt = (j == 7) ? pr : pcv[i + 1][j + 1];
                const float lap = pcv[i][j] + pcv[i + 2][j] + lft + rgt - 4.0f * c;
                pn[i][j] = 2.0f * c - pp[i][j] + c2r[i][j] * lap;
                pp[i][j] = c;                    // old p_cur becomes p_prev
            }
        }
        __syncthreads();                          // all p_cur reads done

        #pragma unroll
        for (int i = 0; i < 4; ++i) {
            float* prow = &P[(lr0 + i) * STR + lc0];
            *(float4*)(prow)     = make_float4(pn[i][0], pn[i][1], pn[i][2], pn[i][3]);
            *(float4*)(prow + 4) = make_float4(pn[i][4], pn[i][5], pn[i][6], pn[i][7]);
        }
        __syncthreads();                          // p_next visible

        // ---- source injection (one lane) ----
        if (own_src && tid == 0) {
            int t2 = t + 2; if (t2 > NT - 1) t2 = NT - 1;
            const float amp = x[t2 * NS + shot] + x[t * NS + shot];
            P[(sz - row0 + 1) * STR + sx + COFF] += amp;
        }
        __syncthreads();                          // injection visible

        // ---- receiver gather (post-injection p_next) ----
        if (rz >= row0 && rz < row0 + HROWS) {
            out[(t * NS + shot) * NREC + tid] = P[(rz - row0 + 1) * STR + rx + COFF];
        }

        // ---- halo exchange with partner workgroup (double-buffered) ----
        float*       sendbuf = halo + (((shot * 2 + me)       * 2 + (t & 1))) * NXX;
        const float* recvbuf = halo + (((shot * 2 + (1 - me)) * 2 + (t & 1))) * NXX;

        if (tid < 64) {
            float4 v = *(const float4*)&P[sendlr * STR + COFF + tid * 4];
            *(float4*)(sendbuf + tid * 4) = v;
        }
        __threadfence();                          // stores reach device scope
        __syncthreads();                          // all sends issued+fenced

        if (tid == 0) {
            __hip_atomic_fetch_add(myflag, 1, __ATOMIC_RELEASE, __HIP_MEMORY_SCOPE_AGENT);
            while (__hip_atomic_load(pflag, __ATOMIC_ACQUIRE, __HIP_MEMORY_SCOPE_AGENT) < t + 1)
                __builtin_amdgcn_s_sleep(1);
        }
        __syncthreads();                          // partner's row is in L2

        if (tid < 64) {
#if HAVE_ASYNC_LDS
            // CDNA5 async global->LDS copy (ASYNCcnt path).
            __builtin_amdgcn_global_load_async_to_lds_b128(
                (gv4i_ptr)(recvbuf + tid * 4),
                (lv4i_ptr)&P[recvlr * STR + COFF + tid * 4],
                0, 0);
#  if __has_builtin(__builtin_amdgcn_s_wait_asynccnt)
            __builtin_amdgcn_s_wait_asynccnt(0);
#  else
            asm volatile("s_wait_asynccnt 0" ::: "memory");
#  endif
#else
            float4 v = *(const float4*)(recvbuf + tid * 4);
            *(float4*)&P[recvlr * STR + COFF + tid * 4] = v;
#endif
        }
        __syncthreads();                          // halo row usable next step
    }
}

extern "C" void kernel_launch(void* const* d_in, const int* in_sizes, int n_in,
                              void* d_out, int out_size, void* d_ws, size_t ws_size,
                              hipStream_t stream) {
    const float* x   = (const float*)d_in[0];   // (500, 8)
    const float* vp  = (const float*)d_in[1];   // (256, 256)
    const int*   src = (const int*)  d_in[2];   // (8, 2)
    const int*   rec = (const int*)  d_in[3];   // (8, 128, 2)
    float* out = (float*)d_out;                 // (500, 8, 128)

    float* halo  = (float*)d_ws;                              // 8*2*2*256 floats = 32 KB
    int*   flags = (int*)((char*)d_ws + (size_t)NS * 2 * 2 * NXX * sizeof(float));

    wave_init_flags<<<1, 32, 0, stream>>>(flags);             // flags must reset every call

    const size_t shbytes = (size_t)LROWS * STR * sizeof(float);   // ~134 KB dynamic LDS
    wave_persistent<<<NS * 2, 1024, shbytes, stream>>>(x, vp, src, rec, out, halo, flags);
}